// MultiHeadSelfAttentionLayer_65051574665404
// MI455X (gfx1250) — compile-verified
//
#include <hip/hip_runtime.h>
#include <math.h>

// MI455X / gfx1250: wave32, WMMA 16x16x4 F32 (fp32 in/out, fp32 accum),
// async global->LDS staging (ASYNCcnt).
typedef __attribute__((ext_vector_type(2))) float v2f;
typedef __attribute__((ext_vector_type(8))) float v8f;

#define B_ 4
#define S_ 2048
#define D_ 1024
#define H_ 16
#define DH_ 64
#define MTOT (B_ * S_)                 // 8192 rows
#define QKV_ELEMS ((size_t)MTOT * D_)  // 8,388,608 floats per matrix

static __device__ __forceinline__ v8f wmma_f32(v2f a, v2f b, v8f c) {
  // D = A(16x4,f32) * B(4x16,f32) + C(16x16,f32)
  return __builtin_amdgcn_wmma_f32_16x16x4_f32(
      /*neg_a=*/false, a, /*neg_b=*/false, b,
      /*c_mod=*/(short)0, c, /*reuse_a=*/false, /*reuse_b=*/false);
}

// Per-lane async copy of 16B global -> LDS (CDNA5, tracked by ASYNCcnt).
// LDS aperture: flat addr[31:0] == wave-relative LDS offset.
static __device__ __forceinline__ void async_load_b128(void* lds_ptr,
                                                       const void* gptr) {
  unsigned loff = (unsigned)(uintptr_t)lds_ptr;
  asm volatile("global_load_async_to_lds_b128 %0, %1, off" ::"v"(loff),
               "v"(gptr)
               : "memory");
}
static __device__ __forceinline__ void wait_async0() {
  asm volatile("s_wait_asynccnt 0x0" ::: "memory");
}

// ---------------------------------------------------------------------------
// Kernel 1: fused QKV projection GEMM.
//   Q/K/V[m][e] = sum_d X[m][d] * W[d][e],  M=8192, N=1024, K=1024.
// Block = 8 waves sharing one 64-wide N group; W staged through LDS in
// 64(k) x 64(n) chunks via async B128 loads (16 KB). One wave -> 16x64 tile.
// ---------------------------------------------------------------------------
__global__ __launch_bounds__(256) void qkv_gemm_kernel(
    const float* __restrict__ X, const float* __restrict__ Wq,
    const float* __restrict__ Wk, const float* __restrict__ Wv,
    float* __restrict__ ws) {
  __shared__ float wlds[64 * 64];  // 16 KB: W[kc..kc+63][n0..n0+63] row-major
  const int tid   = threadIdx.x;
  const int lane  = tid & 31;
  const int wave  = tid >> 5;
  const int lm    = lane & 15;
  const int khalf = (lane >> 4) << 1;        // 0 or 2
  const int mt    = blockIdx.x * 8 + wave;   // 0..511
  const int n0    = blockIdx.y * 64;         // 0..960
  const int which = blockIdx.z;              // 0=Q,1=K,2=V
  const float* W  = (which == 0) ? Wq : (which == 1) ? Wk : Wv;
  float* O        = ws + (size_t)which * QKV_ELEMS;
  const int m0    = mt * 16;
  const float* arow = X + (size_t)(m0 + lm) * D_;
  const int lrow = tid >> 4;         // 0..15 (chunk-relative, +16*i)
  const int lcol = (tid & 15) * 4;   // 0..60

  v8f c[4] = {v8f{}, v8f{}, v8f{}, v8f{}};
  for (int kc = 0; kc < D_; kc += 64) {
    __syncthreads();  // previous chunk fully consumed
#pragma unroll
    for (int i = 0; i < 4; ++i) {
      const int row = lrow + i * 16;
      async_load_b128(&wlds[row * 64 + lcol],
                      W + (size_t)(kc + row) * D_ + n0 + lcol);
    }
    wait_async0();
    __syncthreads();  // chunk visible to all waves
#pragma unroll
    for (int ks = 0; ks < 16; ++ks) {
      const int kk = kc + ks * 4 + khalf;
      v2f a = *(const v2f*)(arow + kk);
      const int kb = (ks * 4 + khalf) * 64;
#pragma unroll
      for (int t = 0; t < 4; ++t) {
        // B fragment: VGPR j, lane L: W[k+ j][n0 + t*16 + lm]
        v2f bf = {wlds[kb + t * 16 + lm], wlds[kb + 64 + t * 16 + lm]};
        c[t] = wmma_f32(a, bf, c[t]);
      }
    }
  }
  // C/D layout: VGPR r, lanes 0-15 -> row m0+r, lanes 16-31 -> row m0+8+r.
  const int rbase = (lane >> 4) << 3;
#pragma unroll
  for (int r = 0; r < 8; ++r) {
    float* op = O + (size_t)(m0 + rbase + r) * D_ + n0 + lm;
    op[0]  = c[0][r];
    op[16] = c[1][r];
    op[32] = c[2][r];
    op[48] = c[3][r];
  }
}

// ---------------------------------------------------------------------------
// Kernel 2: flash attention per (b, h, 16-row q tile). scale = 1/H.
// All 8 waves of a block share one (b,h): K/V 16x64 tiles staged once per
// block via async B128 loads; P transposed C/D->A layout through LDS.
// ---------------------------------------------------------------------------
__global__ __launch_bounds__(256) void attn_kernel(const float* __restrict__ ws,
                                                   float* __restrict__ out) {
  __shared__ float kt[16 * 64];   // 4 KB K tile, row-major [key][dh]
  __shared__ float vt[16 * 64];   // 4 KB V tile, row-major [key][dh]
  __shared__ float pt[8][256];    // 8 KB: per-wave 16x16 P transpose tile
  const float* Q = ws;
  const float* K = ws + QKV_ELEMS;
  const float* V = ws + 2 * QKV_ELEMS;
  const int tid  = threadIdx.x;
  const int lane = tid & 31;
  const int wave = tid >> 5;
  const int lm   = lane & 15;
  const int lh   = lane >> 4;
  const int gw   = blockIdx.x * 8 + wave;  // 0..8191
  const int qt   = gw & 127;
  const int h    = (gw >> 7) & 15;   // uniform across block (8 | 128)
  const int b    = gw >> 11;         // uniform across block
  const int q0   = qt * 16;
  const float scale = 1.0f / (float)H_;
  const int lrow = tid >> 4;        // loader row 0..15
  const int lcol = (tid & 15) * 4;  // loader col 0..60

  // Preload all 16 A-fragments of the 16x64 Q tile (32 VGPRs).
  v2f qa[16];
  const float* qrow = Q + (size_t)(b * S_ + q0 + lm) * D_ + h * DH_;
#pragma unroll
  for (int ks = 0; ks < 16; ++ks) qa[ks] = *(const v2f*)(qrow + ks * 4 + 2 * lh);

  float m[8], l[8], fac[8];
  v8f o[4] = {v8f{}, v8f{}, v8f{}, v8f{}};
#pragma unroll
  for (int r = 0; r < 8; ++r) { m[r] = -INFINITY; l[r] = 0.0f; }

  for (int j = 0; j < 128; ++j) {
    const int k0 = j * 16;
    __syncthreads();  // previous kt/vt/pt consumers done
    async_load_b128(&kt[lrow * 64 + lcol],
                    K + (size_t)(b * S_ + k0 + lrow) * D_ + h * DH_ + lcol);
    async_load_b128(&vt[lrow * 64 + lcol],
                    V + (size_t)(b * S_ + k0 + lrow) * D_ + h * DH_ + lcol);
    wait_async0();
    __syncthreads();  // K/V tiles visible block-wide

    // --- S = Q * K^T : B[k=dh][n=key] = kt[key=lm][dh] ---
    v8f s = {};
#pragma unroll
    for (int ks = 0; ks < 16; ++ks) {
      v2f bf = *(const v2f*)&kt[lm * 64 + ks * 4 + 2 * lh];
      s = wmma_f32(qa[ks], bf, s);
    }
    // --- online softmax: per-row reductions across 16-lane halves ---
#pragma unroll
    for (int r = 0; r < 8; ++r) {
      float x  = s[r] * scale;
      float mx = x;
      mx = fmaxf(mx, __shfl_xor(mx, 1, 32));
      mx = fmaxf(mx, __shfl_xor(mx, 2, 32));
      mx = fmaxf(mx, __shfl_xor(mx, 4, 32));
      mx = fmaxf(mx, __shfl_xor(mx, 8, 32));
      float mn  = fmaxf(m[r], mx);
      float p   = expf(x - mn);
      float sum = p;
      sum += __shfl_xor(sum, 1, 32);
      sum += __shfl_xor(sum, 2, 32);
      sum += __shfl_xor(sum, 4, 32);
      sum += __shfl_xor(sum, 8, 32);
      float f = expf(m[r] - mn);
      l[r]    = l[r] * f + sum;
      m[r]    = mn;
      fac[r]  = f;
      // D-layout element (row r+8*lh, col lm) -> row-major LDS tile
      pt[wave][(r + 8 * lh) * 16 + lm] = p;
    }
#pragma unroll
    for (int r = 0; r < 8; ++r) {
      o[0][r] *= fac[r];
      o[1][r] *= fac[r];
      o[2][r] *= fac[r];
      o[3][r] *= fac[r];
    }
    __syncthreads();  // P tiles written (uniform trip count block-wide)
    v2f pa[4];
#pragma unroll
    for (int ks = 0; ks < 4; ++ks)
      pa[ks] = *(const v2f*)&pt[wave][lm * 16 + ks * 4 + 2 * lh];
    // --- O += P * V : B[k=key_off][n=dh] = vt[key_off][dh] ---
#pragma unroll
    for (int t = 0; t < 4; ++t) {
#pragma unroll
      for (int ks = 0; ks < 4; ++ks) {
        const int kk = ks * 4 + 2 * lh;
        v2f bf = {vt[kk * 64 + t * 16 + lm], vt[(kk + 1) * 64 + t * 16 + lm]};
        o[t] = wmma_f32(pa[ks], bf, o[t]);
      }
    }
  }

  // Epilogue: out[b][q][h*64 + dh] = O / l
#pragma unroll
  for (int r = 0; r < 8; ++r) {
    float inv = 1.0f / l[r];
    int orow  = q0 + r + 8 * lh;
    float* op = out + (size_t)(b * S_ + orow) * D_ + h * DH_ + lm;
    op[0]  = o[0][r] * inv;
    op[16] = o[1][r] * inv;
    op[32] = o[2][r] * inv;
    op[48] = o[3][r] * inv;
  }
}

extern "C" void kernel_launch(void* const* d_in, const int* in_sizes, int n_in,
                              void* d_out, int out_size, void* d_ws,
                              size_t ws_size, hipStream_t stream) {
  const float* X  = (const float*)d_in[0];
  const float* Wq = (const float*)d_in[1];
  const float* Wk = (const float*)d_in[2];
  const float* Wv = (const float*)d_in[3];
  float* ws  = (float*)d_ws;  // needs 3 * 8192 * 1024 * 4B = 96 MB
  float* out = (float*)d_out;

  dim3 g1(64, 16, 3);  // 8 m-tiles/block x 16 n-groups x {Q,K,V}
  qkv_gemm_kernel<<<g1, 256, 0, stream>>>(X, Wq, Wk, Wv, ws);
  attn_kernel<<<1024, 256, 0, stream>>>(ws, out);  // 8192 waves = B*H*S/16
}